// GraphAttentionLayer_61924838474175
// MI455X (gfx1250) — compile-verified
//
#include <hip/hip_runtime.h>
#include <hip/hip_bf16.h>

// GAT layer: N=4096, IN_F=256, OUT_F=64, H=8
#define N_NODES 4096
#define IN_F    256
#define OUT_F   64
#define NHEAD   8
#define HID     512           // H*OUT_F
#define NEG_BIG (-9.0e15f)
#define LN_EPS  1e-05f

typedef __attribute__((ext_vector_type(2))) float v2f;
typedef __attribute__((ext_vector_type(8))) float v8f;

__device__ __forceinline__ v8f wmma_f32_16x16x4(v2f A, v2f B, v8f C) {
    // 8 args: (neg_a, A, neg_b, B, c_mod, C, reuse_a, reuse_b)
    return __builtin_amdgcn_wmma_f32_16x16x4_f32(false, A, false, B, (short)0, C,
                                                 false, false);
}

// ---------------------------------------------------------------------------
// K1: ht[4096,512] = h[4096,256] @ W[256,512]   (f32 WMMA 16x16x4)
// grid (256,8), block 128 (4 waves). Wave w computes 16x16 tile at col w*16.
// ---------------------------------------------------------------------------
__global__ void __launch_bounds__(128)
k1_gemm_ht(const float* __restrict__ h, const float* __restrict__ W,
           float* __restrict__ ht) {
    __shared__ float sA[16 * 260];   // 16 rows x 256 K, padded stride
    const int i0 = blockIdx.x * 16;
    const int n0 = blockIdx.y * 64;
    const int tid  = threadIdx.x;
    const int lane = tid & 31;
    const int wave = tid >> 5;

    for (int idx = tid; idx < 16 * 256 / 4; idx += 128) {
        const int r = idx >> 6, c = (idx & 63) << 2;
        const float4 v = *(const float4*)&h[(size_t)(i0 + r) * IN_F + c];
        *(float4*)&sA[r * 260 + c] = v;   // 260%4==0 -> 16B aligned
    }
    __syncthreads();

    const int m  = lane & 15;   // row within A-tile / col within B-tile
    const int hi = lane >> 4;   // K-half selector

    v8f acc = {};
    for (int ks = 0; ks < IN_F / 4; ++ks) {
        v2f A, B;
#pragma unroll
        for (int v = 0; v < 2; ++v) {
            const int kk = (hi << 1) + v;     // K within step: 0..3
            const int k  = ks * 4 + kk;
            A[v] = sA[m * 260 + k];
            B[v] = W[(size_t)k * HID + n0 + (wave << 4) + m];
        }
        acc = wmma_f32_16x16x4(A, B, acc);
    }

#pragma unroll
    for (int p = 0; p < 8; ++p) {
        const int row = p + (hi << 3);        // C/D layout: m = p + 8*(lane/16)
        ht[(size_t)(i0 + row) * HID + n0 + (wave << 4) + m] = acc[p];
    }
}

// ---------------------------------------------------------------------------
// K2: s_src[n,h] = ht[n,h,:] . a[:64] ; s_dst[n,h] = ht[n,h,:] . a[64:]
// one thread per (n,h) pair -> 32768 threads
// ---------------------------------------------------------------------------
__global__ void __launch_bounds__(256)
k2_scores(const float* __restrict__ ht, const float* __restrict__ a,
          float* __restrict__ ssrc, float* __restrict__ sdst) {
    const int p = blockIdx.x * blockDim.x + threadIdx.x;   // n*8 + h
    if (p >= N_NODES * NHEAD) return;
    const float* row = ht + (size_t)p * OUT_F;
    float s1 = 0.f, s2 = 0.f;
#pragma unroll 8
    for (int f = 0; f < OUT_F; ++f) {
        const float v = row[f];
        s1 += v * a[f];
        s2 += v * a[OUT_F + f];
    }
    ssrc[p] = s1;
    sdst[p] = s2;
}

// ---------------------------------------------------------------------------
// K3: per row i, per head h: M = max_j masked_e, L = sum_j exp(masked_e - M)
// one block (256 thr) per row; two passes over the 16KB adj row (L2-hot 2nd).
// ---------------------------------------------------------------------------
__global__ void __launch_bounds__(256)
k3_rowstats(const int* __restrict__ adj, const float* __restrict__ ssrc,
            const float* __restrict__ sdst, float* __restrict__ Mb,
            float* __restrict__ Lb) {
    __shared__ float red[NHEAD][256];
    const int i   = blockIdx.x;
    const int tid = threadIdx.x;

    float sr[NHEAD];
#pragma unroll
    for (int hh = 0; hh < NHEAD; ++hh) sr[hh] = ssrc[i * NHEAD + hh];

    // pass 1: masked row max
    float mx[NHEAD];
#pragma unroll
    for (int hh = 0; hh < NHEAD; ++hh) mx[hh] = NEG_BIG;
    for (int j = tid; j < N_NODES; j += 256) {
        const int av = adj[(size_t)i * N_NODES + j];
#pragma unroll
        for (int hh = 0; hh < NHEAD; ++hh) {
            float e = sr[hh] + sdst[j * NHEAD + hh];
            e = e > 0.f ? e : 0.2f * e;               // leaky_relu(0.2)
            const float val = av > 0 ? e : NEG_BIG;
            mx[hh] = fmaxf(mx[hh], val);
        }
    }
#pragma unroll
    for (int hh = 0; hh < NHEAD; ++hh) red[hh][tid] = mx[hh];
    __syncthreads();
    for (int s = 128; s >= 1; s >>= 1) {
        if (tid < s) {
#pragma unroll
            for (int hh = 0; hh < NHEAD; ++hh)
                red[hh][tid] = fmaxf(red[hh][tid], red[hh][tid + s]);
        }
        __syncthreads();
    }
    float Mh[NHEAD];
#pragma unroll
    for (int hh = 0; hh < NHEAD; ++hh) Mh[hh] = red[hh][0];
    __syncthreads();

    // pass 2: sum of exp(val - M)
    float sm[NHEAD];
#pragma unroll
    for (int hh = 0; hh < NHEAD; ++hh) sm[hh] = 0.f;
    for (int j = tid; j < N_NODES; j += 256) {
        const int av = adj[(size_t)i * N_NODES + j];
#pragma unroll
        for (int hh = 0; hh < NHEAD; ++hh) {
            float e = sr[hh] + sdst[j * NHEAD + hh];
            e = e > 0.f ? e : 0.2f * e;
            const float val = av > 0 ? e : NEG_BIG;
            sm[hh] += __expf(val - Mh[hh]);
        }
    }
#pragma unroll
    for (int hh = 0; hh < NHEAD; ++hh) red[hh][tid] = sm[hh];
    __syncthreads();
    for (int s = 128; s >= 1; s >>= 1) {
        if (tid < s) {
#pragma unroll
            for (int hh = 0; hh < NHEAD; ++hh)
                red[hh][tid] += red[hh][tid + s];
        }
        __syncthreads();
    }
    if (tid < NHEAD) {
        Mb[i * NHEAD + tid] = Mh[tid];
        Lb[i * NHEAD + tid] = red[tid][0];
    }
}

// ---------------------------------------------------------------------------
// K4 (fused): attention aggregation (WMMA) + head mean + LayerNorm.
// grid 256 (16-row tiles), block 128 (4 waves; wave w owns feature cols
// [16w,16w+16) of each head). Per-head accumulators live in VGPRs (forced
// unroll of the head loop). ht tile stored transposed [f][j] with pad-68
// stride so B-operand LDS reads merge to conflict-free ds_load_b64; adj tile
// pad-68 likewise for the A-operand pair.
// ---------------------------------------------------------------------------
#define JT 64
__global__ void __launch_bounds__(128)
k4_agg_ln(const int* __restrict__ adj, const float* __restrict__ ht,
          const float* __restrict__ ssrc, const float* __restrict__ sdst,
          const float* __restrict__ Mb, const float* __restrict__ Lb,
          const float* __restrict__ gamma, const float* __restrict__ beta,
          float* __restrict__ out) {
    __shared__ int   s_adj[16 * 68];       // adj tile, pad-68 (aligned + no conflicts)
    __shared__ float s_sdst[JT * NHEAD];   // s_dst for chunk, all heads
    __shared__ float s_htT[OUT_F * 68];    // ht tile for one head, TRANSPOSED [f][j]
    __shared__ float s_out[16 * OUT_F];    // hp tile for LayerNorm
    __shared__ float s_L[16 * NHEAD];
    __shared__ float s_M[16 * NHEAD];
    __shared__ float s_ss[16 * NHEAD];
    __shared__ float s_mu[16], s_rs[16];

    const int i0   = blockIdx.x * 16;
    const int tid  = threadIdx.x;
    const int lane = tid & 31;
    const int wave = tid >> 5;
    const int m    = lane & 15;
    const int hi   = lane >> 4;
    const int fcol = (wave << 4) + m;      // feature column owned by this lane

    // per-row stats for this tile (128 entries = blockDim)
    {
        const int r = tid >> 3, hh = tid & 7;
        s_L[r * NHEAD + hh]  = Lb[(i0 + r) * NHEAD + hh];
        s_M[r * NHEAD + hh]  = Mb[(i0 + r) * NHEAD + hh];
        s_ss[r * NHEAD + hh] = ssrc[(i0 + r) * NHEAD + hh];
    }
    __syncthreads();

    v8f acc[NHEAD] = {};

    for (int j0 = 0; j0 < N_NODES; j0 += JT) {
        // prefetch next adj tile rows into cache
        if (j0 + JT < N_NODES) {
            const int r = tid >> 3;
            if (r < 16)
                __builtin_prefetch(&adj[(size_t)(i0 + r) * N_NODES + j0 + JT], 0, 1);
        }
        // stage adj tile (int4 loads, aligned int4 LDS stores: 68%4==0)
        for (int idx = tid; idx < 16 * JT / 4; idx += 128) {
            const int r = idx >> 4, c = (idx & 15) << 2;
            const int4 v = *(const int4*)&adj[(size_t)(i0 + r) * N_NODES + j0 + c];
            *(int4*)&s_adj[r * 68 + c] = v;
        }
        // stage s_dst chunk (contiguous 512 floats -> one float4 per thread)
        {
            const float4 v = *(const float4*)&sdst[j0 * NHEAD + (tid << 2)];
            *(float4*)&s_sdst[tid << 2] = v;
        }
        __syncthreads();

#pragma unroll
        for (int hh = 0; hh < NHEAD; ++hh) {
            // stage ht tile for this head, transposed into [f][j]
            for (int idx = tid; idx < JT * OUT_F / 4; idx += 128) {
                const int jl = idx >> 4, f4 = (idx & 15) << 2;
                const float4 v =
                    *(const float4*)&ht[(size_t)(j0 + jl) * HID + hh * OUT_F + f4];
                s_htT[(f4 + 0) * 68 + jl] = v.x;
                s_htT[(f4 + 1) * 68 + jl] = v.y;
                s_htT[(f4 + 2) * 68 + jl] = v.z;
                s_htT[(f4 + 3) * 68 + jl] = v.w;
            }
            __syncthreads();

            const float ssr  = s_ss[m * NHEAD + hh];
            const float Mrow = s_M[m * NHEAD + hh];
            v8f a_acc = acc[hh];
            for (int ks = 0; ks < JT / 4; ++ks) {
                const int jbase = ks * 4 + (hi << 1);   // even -> b64-aligned pairs
                v2f A, B;
#pragma unroll
                for (int v = 0; v < 2; ++v) {
                    const int jl = jbase + v;
                    // A: softmax numerator P[i, j]
                    const int adjv = s_adj[m * 68 + jl];
                    float e = ssr + s_sdst[jl * NHEAD + hh];
                    e = e > 0.f ? e : 0.2f * e;
                    const float val = adjv > 0 ? e : NEG_BIG;
                    A[v] = __expf(val - Mrow);
                    // B: ht[j, f] from transposed tile (adjacent jl -> b64)
                    B[v] = s_htT[fcol * 68 + jl];
                }
                a_acc = wmma_f32_16x16x4(A, B, a_acc);
            }
            acc[hh] = a_acc;
            __syncthreads();   // before next head overwrites s_htT
        }
        __syncthreads();       // before next chunk overwrites s_adj/s_sdst
    }

    // normalize by L, average heads -> hp tile, then LayerNorm
    v8f hp = {};
#pragma unroll
    for (int hh = 0; hh < NHEAD; ++hh) {
#pragma unroll
        for (int p = 0; p < 8; ++p) {
            const int row = p + (hi << 3);
            hp[p] += acc[hh][p] / s_L[row * NHEAD + hh];
        }
    }
#pragma unroll
    for (int p = 0; p < 8; ++p) {
        const int row = p + (hi << 3);
        s_out[row * OUT_F + fcol] = hp[p] * (1.0f / NHEAD);
    }
    __syncthreads();

    if (tid < 16) {
        float mu = 0.f;
        for (int f = 0; f < OUT_F; ++f) mu += s_out[tid * OUT_F + f];
        mu *= (1.0f / OUT_F);
        float var = 0.f;
        for (int f = 0; f < OUT_F; ++f) {
            const float d = s_out[tid * OUT_F + f] - mu;
            var += d * d;
        }
        var *= (1.0f / OUT_F);
        s_mu[tid] = mu;
        s_rs[tid] = rsqrtf(var + LN_EPS);
    }
    __syncthreads();

    for (int idx = tid; idx < 16 * OUT_F; idx += 128) {
        const int r = idx >> 6, f = idx & 63;
        out[(size_t)(i0 + r) * OUT_F + f] =
            (s_out[r * OUT_F + f] - s_mu[r]) * s_rs[r] * gamma[f] + beta[f];
    }
}

// ---------------------------------------------------------------------------
extern "C" void kernel_launch(void* const* d_in, const int* in_sizes, int n_in,
                              void* d_out, int out_size, void* d_ws, size_t ws_size,
                              hipStream_t stream) {
    const float* h     = (const float*)d_in[0];
    const int*   adj   = (const int*)  d_in[1];
    const float* W     = (const float*)d_in[2];
    const float* a     = (const float*)d_in[3];
    const float* gamma = (const float*)d_in[4];
    const float* beta  = (const float*)d_in[5];
    float* out = (float*)d_out;

    float* ws   = (float*)d_ws;
    float* ht   = ws;                                   // 4096*512 = 2,097,152
    float* ssrc = ht + (size_t)N_NODES * HID;           // 32768
    float* sdst = ssrc + N_NODES * NHEAD;               // 32768
    float* Mb   = sdst + N_NODES * NHEAD;               // 32768
    float* Lb   = Mb   + N_NODES * NHEAD;               // 32768

    k1_gemm_ht<<<dim3(N_NODES / 16, HID / 64), 128, 0, stream>>>(h, W, ht);
    k2_scores<<<(N_NODES * NHEAD) / 256, 256, 0, stream>>>(ht, a, ssrc, sdst);
    k3_rowstats<<<N_NODES, 256, 0, stream>>>(adj, ssrc, sdst, Mb, Lb);
    k4_agg_ln<<<N_NODES / 16, 128, 0, stream>>>(adj, ht, ssrc, sdst, Mb, Lb,
                                                gamma, beta, out);
}